// MultiHeadAttentionSelf_69681549410313
// MI455X (gfx1250) — compile-verified
//
#include <hip/hip_runtime.h>

// MI455X (gfx1250) fused decode-step MHA.
// HBM-bound (~490MB moved); KV clone fused with attention so each cache
// element is read exactly once. GEMMs use native-f32 WMMA 16x16x4.
// Streaming KV traffic uses non-temporal hints (th:TH_NT) so the 235MB
// one-shot stream does not evict the reusable working set from the 192MB L2.

typedef __attribute__((ext_vector_type(2))) float v2f;
typedef __attribute__((ext_vector_type(4))) float v4f;
typedef __attribute__((ext_vector_type(8))) float v8f;

#define DD 1024
#define SS 448
#define BB 64
#define HH 16

// ---------------------------------------------------------------------------
// C[64x1024] = A[64x1024] @ W[1024x1024] + bias  (bias may be null)
// One wave per 16x16 tile; V_WMMA_F32_16X16X4_F32, K stepped by 4.
// A frag layout (ISA 7.12.2, 32-bit A 16x4): lane = M%16 + 16*(K>>1), vgpr=K&1.
// B frag mirrored: lane = N%16 + 16*(K>>1), vgpr=K&1.
// C/D layout: vgpr r -> M = r + 8*(lane>>4), N = lane&15.
// ---------------------------------------------------------------------------
__global__ __launch_bounds__(256) void gemm64x1024_wmma_f32(
    const float* __restrict__ A, const float* __restrict__ W,
    const float* __restrict__ bias, float* __restrict__ C)
{
  const int lane = threadIdx.x & 31;
  const int wave = threadIdx.x >> 5;
  const int n0   = (blockIdx.x * 8 + wave) * 16;
  const int m0   = blockIdx.y * 16;
  const int half = lane >> 4;   // 0 or 1
  const int l16  = lane & 15;

  v8f c = {};
  const float* Arow = A + (size_t)(m0 + l16) * DD;
  const int    n    = n0 + l16;

  for (int kk = 0; kk < DD; kk += 4) {
    const int kA = kk + (half << 1);
    v2f a, b;
    a.x = Arow[kA];
    a.y = Arow[kA + 1];
    const float* Wp = W + (size_t)kA * DD + n;
    b.x = Wp[0];
    b.y = Wp[DD];
    c = __builtin_amdgcn_wmma_f32_16x16x4_f32(false, a, false, b,
                                              (short)0, c, false, false);
  }

  const float bv = bias ? bias[n] : 0.0f;
#pragma unroll
  for (int r = 0; r < 8; ++r) {
    const int m = m0 + (half << 3) + r;
    C[(size_t)m * DD + n] = c[r] + bv;
  }
}

// ---------------------------------------------------------------------------
// Fused: k_upd = scatter(k_cache, k_new @ idx); scores[b,h,s] = 0.125*q.k + mask
// One wave per (b, s) row. Reads K row once (b128 NT, coalesced), writes k_upd
// (b128 NT), reduces per-head dot products with 16-lane butterflies.
// ---------------------------------------------------------------------------
__global__ __launch_bounds__(256) void k_copy_score(
    const float* __restrict__ Kc, const float* __restrict__ Knew,
    const float* __restrict__ Q, const float* __restrict__ mask,
    const long long* __restrict__ curidx,
    float* __restrict__ Kupd, float* __restrict__ scores)
{
  __shared__ float qs[DD];
  const int b = blockIdx.y;
  for (int i = threadIdx.x; i < DD; i += 256) qs[i] = Q[(size_t)b * DD + i];
  __syncthreads();

  const int idx  = (int)curidx[0];
  const int lane = threadIdx.x & 31;
  const int wave = threadIdx.x >> 5;
  const int s    = blockIdx.x * 8 + wave;           // 56*8 == 448

  const float* krow = (s == idx) ? (Knew + (size_t)b * DD)
                                 : (Kc + ((size_t)b * SS + s) * DD);
  float* kout = Kupd + ((size_t)b * SS + s) * DD;
  const float mk = mask[s];

#pragma unroll
  for (int j = 0; j < 8; ++j) {
    const int off = (j * 32 + lane) * 4;            // head = (j*32+lane)>>4
    const v4f kv = __builtin_nontemporal_load((const v4f*)(krow + off));
    __builtin_nontemporal_store(kv, (v4f*)(kout + off));
    const v4f qv = *(const v4f*)(qs + off);
    float p = kv.x * qv.x + kv.y * qv.y + kv.z * qv.z + kv.w * qv.w;
    p += __shfl_xor(p, 1, 16);
    p += __shfl_xor(p, 2, 16);
    p += __shfl_xor(p, 4, 16);
    p += __shfl_xor(p, 8, 16);
    if ((lane & 15) == 0) {
      const int h = j * 2 + (lane >> 4);
      scores[((size_t)b * HH + h) * SS + s] = p * 0.125f + mk;
    }
  }
}

// ---------------------------------------------------------------------------
// In-place softmax over S=448 per (b,h) row. One wave per row, 14 elems/lane.
// ---------------------------------------------------------------------------
__global__ __launch_bounds__(256) void softmax448(float* __restrict__ w)
{
  const int lane = threadIdx.x & 31;
  const int wave = threadIdx.x >> 5;
  const int row  = blockIdx.x * 8 + wave;           // 0..1023
  float* p = w + (size_t)row * SS;

  float mx = -3.4e38f;
  float v[14];
  int c = 0;
  for (int i = lane; i < SS; i += 32) { v[c] = p[i]; mx = fmaxf(mx, v[c]); ++c; }
#pragma unroll
  for (int d = 16; d >= 1; d >>= 1) mx = fmaxf(mx, __shfl_xor(mx, d, 32));

  float sum = 0.0f;
#pragma unroll
  for (int i = 0; i < 14; ++i) { v[i] = __expf(v[i] - mx); sum += v[i]; }
#pragma unroll
  for (int d = 16; d >= 1; d >>= 1) sum += __shfl_xor(sum, d, 32);

  const float inv = 1.0f / sum;
  c = 0;
  for (int i = lane; i < SS; i += 32) { p[i] = v[c] * inv; ++c; }
}

// ---------------------------------------------------------------------------
// Fused: v_upd = scatter(v_cache, v_new @ idx); partial[b] += w[b,h,s]*v[s,d]
// Block = (s-chunk of 64, b). Weights staged in LDS. Thread owns a v4f
// column slice; V row read once (b128 NT), stored to v_upd (b128 NT),
// FMA'd into the accumulator.
// ---------------------------------------------------------------------------
__global__ __launch_bounds__(256) void v_copy_acc(
    const float* __restrict__ Vc, const float* __restrict__ Vnew,
    const float* __restrict__ w, const long long* __restrict__ curidx,
    float* __restrict__ Vupd, float* __restrict__ partials)
{
  __shared__ float wl[HH * 64];                     // [h][s_local]
  const int b  = blockIdx.y;
  const int s0 = blockIdx.x * 64;
  for (int i = threadIdx.x; i < HH * 64; i += 256) {
    const int h = i >> 6, sl = i & 63;
    wl[i] = w[((size_t)b * HH + h) * SS + (s0 + sl)];
  }
  __syncthreads();

  const int idx = (int)curidx[0];
  const int d   = threadIdx.x * 4;
  const int h   = threadIdx.x >> 4;                 // (d>>6)

  v4f acc = {};
  for (int sl = 0; sl < 64; ++sl) {
    const int s = s0 + sl;
    const float* vrow = (s == idx) ? (Vnew + (size_t)b * DD)
                                   : (Vc + ((size_t)b * SS + s) * DD);
    const v4f vv = __builtin_nontemporal_load((const v4f*)(vrow + d));
    __builtin_nontemporal_store(vv, (v4f*)(Vupd + ((size_t)b * SS + s) * DD + d));
    const float wgt = wl[(h << 6) + sl];
    acc += wgt * vv;
  }
  *(v4f*)(partials + ((size_t)blockIdx.x * BB + b) * DD + d) = acc;
}

// wv[b,d] = sum over 7 S-chunk partials
__global__ __launch_bounds__(256) void v_reduce(
    const float* __restrict__ partials, float* __restrict__ wv)
{
  const int i = blockIdx.x * 256 + threadIdx.x;     // 0..65535
  float s = 0.0f;
#pragma unroll
  for (int p = 0; p < 7; ++p) s += partials[(size_t)p * (BB * DD) + i];
  wv[i] = s;
}

// ---------------------------------------------------------------------------
extern "C" void kernel_launch(void* const* d_in, const int* in_sizes, int n_in,
                              void* d_out, int out_size, void* d_ws, size_t ws_size,
                              hipStream_t stream) {
  const float*      x    = (const float*)d_in[0];
  const float*      kc   = (const float*)d_in[1];
  const float*      vc   = (const float*)d_in[2];
  const long long*  cidx = (const long long*)d_in[3];
  const float*      mask = (const float*)d_in[4];
  const float*      Wq   = (const float*)d_in[5];
  const float*      bq   = (const float*)d_in[6];
  const float*      Wk   = (const float*)d_in[7];
  const float*      Wv   = (const float*)d_in[8];
  const float*      bvp  = (const float*)d_in[9];
  const float*      Wo   = (const float*)d_in[10];
  const float*      bo   = (const float*)d_in[11];

  float* out  = (float*)d_out;                      // [64,1,1024]
  float* kupd = out + (size_t)BB * DD;              // [64,448,1024]
  float* vupd = kupd + (size_t)BB * SS * DD;        // [64,448,1024]

  float* ws       = (float*)d_ws;
  float* q        = ws;                             // 65536
  float* knew     = q + BB * DD;                    // 65536
  float* vnew     = knew + BB * DD;                 // 65536
  float* scores   = vnew + BB * DD;                 // 64*16*448 = 458752
  float* partials = scores + (size_t)BB * HH * SS;  // 7*65536   = 458752
  float* wv       = partials + (size_t)7 * BB * DD; // 65536

  const dim3 gGemm(8, 4), bGemm(256);
  // q/k/v projections (WMMA f32)
  gemm64x1024_wmma_f32<<<gGemm, bGemm, 0, stream>>>(x, Wq, bq, q);
  gemm64x1024_wmma_f32<<<gGemm, bGemm, 0, stream>>>(x, Wk, nullptr, knew);
  gemm64x1024_wmma_f32<<<gGemm, bGemm, 0, stream>>>(x, Wv, bvp, vnew);

  // fused K scatter-copy + attention scores
  k_copy_score<<<dim3(56, 64), 256, 0, stream>>>(kc, knew, q, mask, cidx,
                                                 kupd, scores);
  // softmax over S, in place
  softmax448<<<128, 256, 0, stream>>>(scores);

  // fused V scatter-copy + weighted accumulate, then reduce
  v_copy_acc<<<dim3(7, 64), 256, 0, stream>>>(vc, vnew, scores, cidx,
                                              vupd, partials);
  v_reduce<<<256, 256, 0, stream>>>(partials, wv);

  // output projection (WMMA f32)
  gemm64x1024_wmma_f32<<<gGemm, bGemm, 0, stream>>>(wv, Wo, bo, out);
}